// MixHop_48163763258019
// MI455X (gfx1250) — compile-verified
//
#include <hip/hip_runtime.h>

#define NNODES 100000
#define NEDGES 1600000
#define INC 128
#define HIDC 128
#define OUTC 64
#define CATC 384
#define EPSV 1e-5f

typedef __attribute__((ext_vector_type(2))) float v2f;
typedef __attribute__((ext_vector_type(8))) float v8f;

// ---------------- degree / gcn_norm ----------------
__global__ void deg_kernel(const long long* __restrict__ col, float* __restrict__ deg) {
  int e = blockIdx.x * blockDim.x + threadIdx.x;   // grid sized exactly E
  atomicAdd(&deg[col[e]], 1.0f);
}

__global__ void dinv_kernel(float* __restrict__ deg) {
  int i = blockIdx.x * blockDim.x + threadIdx.x;
  if (i < NNODES) {
    float d = deg[i];
    deg[i] = d > 0.f ? rsqrtf(d) : 0.f;           // deg>0 => deg>=1 => max(deg,1)=deg
  }
}

__global__ void ew_kernel(const long long* __restrict__ row, const long long* __restrict__ col,
                          const float* __restrict__ dinv, float* __restrict__ ew) {
  int e = blockIdx.x * blockDim.x + threadIdx.x;   // grid sized exactly E
  ew[e] = dinv[row[e]] * dinv[col[e]];
}

// ---------------- SpMM: out[col[e]] += ew[e] * h[row[e]] ----------------
// 32/16 consecutive lanes = the float4 quads of one edge -> edge data is
// wave-uniform (scalarizable), feature traffic fully coalesced, atomics
// resolve in L2 (the whole N x 128 feature block is 51.2 MB < 192 MB L2).
__global__ void spmm_kernel(const float* __restrict__ h, int hstride, int c4,
                            const long long* __restrict__ row, const long long* __restrict__ col,
                            const float* __restrict__ ew,
                            float* __restrict__ out, int ostride) {
  long long t = (long long)blockIdx.x * blockDim.x + threadIdx.x;  // grid sized exactly E*c4
  int e = (int)(t / c4);
  int q = (int)(t % c4);
  long long r = row[e];
  long long c = col[e];
  float w = ew[e];
  const float4 v = *(const float4*)(h + r * (long long)hstride + q * 4);
  float* o = out + c * (long long)ostride + q * 4;
  atomicAdd(o + 0, w * v.x);
  atomicAdd(o + 1, w * v.y);
  atomicAdd(o + 2, w * v.z);
  atomicAdd(o + 3, w * v.w);
}

// ---------------- GEMM via V_WMMA_F32_16X16X4_F32 ----------------
// out[row, ocol+col] = H(N x K) @ W(K x M) + bias, out row-stride = ostride.
// One wave computes a 16x32 tile (two accumulators): the A fragment is loaded
// once per K-step and feeds two back-to-back WMMAs -> halves A traffic vs a
// one-tile-per-wave layout and doubles WMMA density.
// No divergence anywhere: EXEC all ones as required by WMMA.
__global__ void gemm_wmma(const float* __restrict__ H, int K,
                          const float* __restrict__ W, const float* __restrict__ bias, int M,
                          float* __restrict__ out, int ostride, int ocol) {
  const int lane = threadIdx.x & 31;
  const int wave = threadIdx.x >> 5;
  const int row0 = blockIdx.x << 4;     // N/16 exact
  const int col0 = wave << 5;           // 32 columns per wave
  const int half = lane >> 4;           // lanes 16-31 carry K+2/K+3
  const int l    = lane & 15;

  const float* hrow = H + (long long)(row0 + l) * K + 2 * half;       // A: M=l, K pair
  const float* wcol = W + (long long)(2 * half) * M + col0 + l;       // B: N=l, K pair

  v8f acc0 = {};
  v8f acc1 = {};
  for (int k = 0; k < K; k += 4) {
    // pull the next chunk of this activation row toward the WGP ahead of use
    __builtin_prefetch(hrow + k + 32, 0, 1);      // -> global_prefetch_b8
    v2f a, b0, b1;
    a.x = hrow[k];                        // K = k + 2*half
    a.y = hrow[k + 1];                    // K = k + 2*half + 1
    const float* wk = wcol + (long long)k * M;
    b0.x = wk[0];
    b0.y = wk[M];
    b1.x = wk[16];
    b1.y = wk[M + 16];
    acc0 = __builtin_amdgcn_wmma_f32_16x16x4_f32(
        false, a, false, b0, (short)0, acc0, false, false);
    acc1 = __builtin_amdgcn_wmma_f32_16x16x4_f32(
        false, a, false, b1, (short)0, acc1, false, false);
  }

  // C/D layout: VGPR r -> M = r + 8*half, N = l
  float bc0 = bias[col0 + l];
  float bc1 = bias[col0 + 16 + l];
  float* orow = out + (long long)(row0 + 8 * half) * ostride + ocol + col0 + l;
#pragma unroll
  for (int r = 0; r < 8; ++r) {
    orow[(long long)r * ostride]      = acc0[r] + bc0;
    orow[(long long)r * ostride + 16] = acc1[r] + bc1;
  }
}

// ---------------- BatchNorm (training stats) ----------------
__global__ void bn_stats_kernel(const float* __restrict__ h, int C,
                                const float* __restrict__ g, const float* __restrict__ bb,
                                float* __restrict__ scale, float* __restrict__ shift) {
  int c = blockIdx.x;                    // one block per column
  float s = 0.f, sq = 0.f;
  for (int r = threadIdx.x; r < NNODES; r += 256) {
    float v = h[(long long)r * C + c];
    s += v; sq += v * v;
  }
  __shared__ float ss[256];
  __shared__ float ssq[256];
  ss[threadIdx.x] = s; ssq[threadIdx.x] = sq;
  __syncthreads();
  for (int o = 128; o > 0; o >>= 1) {
    if (threadIdx.x < o) {
      ss[threadIdx.x]  += ss[threadIdx.x + o];
      ssq[threadIdx.x] += ssq[threadIdx.x + o];
    }
    __syncthreads();
  }
  if (threadIdx.x == 0) {
    const float inv = 1.f / (float)NNODES;
    float mu  = ss[0] * inv;
    float var = ssq[0] * inv - mu * mu;   // biased variance
    float sc  = g[c] * rsqrtf(var + EPSV);
    scale[c] = sc;
    shift[c] = bb[c] - mu * sc;
  }
}

__global__ void bn_apply_relu_kernel(float* __restrict__ h, int C,
                                     const float* __restrict__ scale,
                                     const float* __restrict__ shift,
                                     long long total) {
  long long t = (long long)blockIdx.x * blockDim.x + threadIdx.x;
  if (t >= total) return;
  int c = (int)(t % C);
  float v = h[t] * scale[c] + shift[c];
  h[t] = v > 0.f ? v : 0.f;
}

// ---------------- host orchestration ----------------
extern "C" void kernel_launch(void* const* d_in, const int* in_sizes, int n_in,
                              void* d_out, int out_size, void* d_ws, size_t ws_size,
                              hipStream_t stream) {
  (void)in_sizes; (void)n_in; (void)out_size; (void)ws_size;
  const float*     x    = (const float*)d_in[0];
  const long long* ei   = (const long long*)d_in[1];
  const long long* row  = ei;
  const long long* col  = ei + NEDGES;
  const float* W0  = (const float*)d_in[2];
  const float* b0  = (const float*)d_in[3];
  const float* W1  = (const float*)d_in[4];
  const float* b1  = (const float*)d_in[5];
  const float* W2  = (const float*)d_in[6];
  const float* b2  = (const float*)d_in[7];
  const float* bn0g = (const float*)d_in[8];
  const float* bn0b = (const float*)d_in[9];
  const float* bn1g = (const float*)d_in[10];
  const float* bn1b = (const float*)d_in[11];
  const float* fpW = (const float*)d_in[12];
  const float* fpb = (const float*)d_in[13];
  float* out = (float*)d_out;

  // workspace carve-out (256B aligned)
  char* p = (char*)d_ws;
  auto carve = [&](size_t bytes) {
    char* q = p;
    p += (bytes + 255) & ~(size_t)255;
    return q;
  };
  float* deg   = (float*)carve((size_t)NNODES * 4);
  float* ew    = (float*)carve((size_t)NEDGES * 4);
  float* tmp1  = (float*)carve((size_t)NNODES * HIDC * 4);
  float* tmp2  = (float*)carve((size_t)NNODES * HIDC * 4);
  float* catA  = (float*)carve((size_t)NNODES * CATC * 4);
  float* catB  = (float*)carve((size_t)NNODES * CATC * 4);
  float* scale = (float*)carve(CATC * 4);
  float* shift = (float*)carve(CATC * 4);

  // gcn_norm edge weights
  hipMemsetAsync(deg, 0, (size_t)NNODES * 4, stream);
  deg_kernel<<<NEDGES / 256, 256, 0, stream>>>(col, deg);
  dinv_kernel<<<(NNODES + 255) / 256, 256, 0, stream>>>(deg);
  ew_kernel<<<NEDGES / 256, 256, 0, stream>>>(row, col, deg, ew);

  auto gemm = [&](const float* H, int K, const float* W, const float* bias, int M,
                  float* o, int ostride, int ocol) {
    gemm_wmma<<<NNODES / 16, 32 * (M / 32), 0, stream>>>(H, K, W, bias, M, o, ostride, ocol);
  };
  auto spmm = [&](const float* h, int hstride, int C, float* o, int ostride) {
    long long total = (long long)NEDGES * (C / 4);   // divisible by 256 for C in {64,128}
    spmm_kernel<<<(int)(total / 256), 256, 0, stream>>>(h, hstride, C / 4, row, col, ew, o, ostride);
  };
  // mixhop: cat[:,0:M]=hW0+b0 ; cat[:,M:2M]=A(hW1+b1) ; cat[:,2M:3M]=A^2(hW2+b2)
  auto mixhop = [&](const float* hin, int K, const float* W, const float* b, int M,
                    float* cat, int cstride) {
    hipMemsetAsync(cat, 0, (size_t)NNODES * cstride * 4, stream);
    gemm(hin, K, W, b, M, cat, cstride, 0);
    gemm(hin, K, W + (size_t)K * M, b + M, M, tmp1, M, 0);
    spmm(tmp1, M, M, cat + M, cstride);
    gemm(hin, K, W + 2 * (size_t)K * M, b + 2 * M, M, tmp1, M, 0);
    hipMemsetAsync(tmp2, 0, (size_t)NNODES * M * 4, stream);
    spmm(tmp1, M, M, tmp2, M);
    spmm(tmp2, M, M, cat + 2 * M, cstride);
  };

  const long long catTot = (long long)NNODES * CATC;  // 38.4M, /256 exact

  // layer 0: x -> catA, bn0 + relu
  mixhop(x, INC, W0, b0, HIDC, catA, CATC);
  bn_stats_kernel<<<CATC, 256, 0, stream>>>(catA, CATC, bn0g, bn0b, scale, shift);
  bn_apply_relu_kernel<<<(int)(catTot / 256), 256, 0, stream>>>(catA, CATC, scale, shift, catTot);

  // layer 1: catA -> catB, bn1 + relu
  mixhop(catA, CATC, W1, b1, HIDC, catB, CATC);
  bn_stats_kernel<<<CATC, 256, 0, stream>>>(catB, CATC, bn1g, bn1b, scale, shift);
  bn_apply_relu_kernel<<<(int)(catTot / 256), 256, 0, stream>>>(catB, CATC, scale, shift, catTot);

  // layer 2: catB -> catA reused as N x 192 (no bn)
  mixhop(catB, CATC, W2, b2, OUTC, catA, 3 * OUTC);

  // final projection: (N x 192) @ (192 x 64) + fpb -> d_out
  gemm(catA, 3 * OUTC, fpW, fpb, OUTC, out, OUTC, 0);
}